// FS_AttPool_55095840473451
// MI455X (gfx1250) — compile-verified
//
#include <hip/hip_runtime.h>
#include <cstdint>
#include <cstddef>

#define NN   2048   // N
#define BB   64     // B
#define DD   512    // D
#define SEL  512    // N / PS

// native 16-byte vector type (HIP's float4 is a struct and is rejected by
// __builtin_nontemporal_load/store)
typedef __attribute__((ext_vector_type(4))) float v4f;

// ---------------------------------------------------------------------------
// Branchless insert of v into descending-sorted top-5 list a[0..4].
// Compare-exchange chain: lowers to v_max_f32/v_min_f32, no divergence.
// ---------------------------------------------------------------------------
__device__ __forceinline__ void ins5(float v, float a[5]) {
    float t;
    t = fminf(a[0], v); a[0] = fmaxf(a[0], v); v = t;
    t = fminf(a[1], v); a[1] = fmaxf(a[1], v); v = t;
    t = fminf(a[2], v); a[2] = fmaxf(a[2], v); v = t;
    t = fminf(a[3], v); a[3] = fmaxf(a[3], v); v = t;
    a[4] = fmaxf(a[4], v);
}

// ---------------------------------------------------------------------------
// Kernel 1: per-row 5th-largest threshold.
// blockIdx.x in [0,128): rows 0..63 -> attention, 64..127 -> att_off
// ---------------------------------------------------------------------------
__global__ __launch_bounds__(256) void thr_kernel(const float* __restrict__ att,
                                                  const float* __restrict__ aoff,
                                                  float* __restrict__ thr) {
    __shared__ float tops[256 * 5];
    const int    r   = blockIdx.x;
    const float* src = (r < BB) ? att : aoff;
    const float* row = src + (size_t)(r & (BB - 1)) * NN;
    const int t = threadIdx.x;

    float a[5] = {-INFINITY, -INFINITY, -INFINITY, -INFINITY, -INFINITY};
    #pragma unroll
    for (int k = 0; k < NN / 256; ++k) ins5(row[t + k * 256], a);
    #pragma unroll
    for (int k = 0; k < 5; ++k) tops[t * 5 + k] = a[k];
    __syncthreads();

    for (int s = 128; s >= 1; s >>= 1) {
        if (t < s) {
            #pragma unroll
            for (int k = 0; k < 5; ++k) a[k] = tops[t * 5 + k];
            #pragma unroll
            for (int k = 0; k < 5; ++k) ins5(tops[(t + s) * 5 + k], a);
            #pragma unroll
            for (int k = 0; k < 5; ++k) tops[t * 5 + k] = a[k];
        }
        __syncthreads();
    }
    if (t == 0) thr[r] = tops[4];   // 5th largest == sorted_desc[PS]
}

// ---------------------------------------------------------------------------
// Kernel 2: M[n] = sum_b 0.5*(att[b,n] >= thrA[b]) + 0.5*(off[b,n] >= thrO[b])
// ---------------------------------------------------------------------------
__global__ __launch_bounds__(256) void m_kernel(const float* __restrict__ att,
                                                const float* __restrict__ aoff,
                                                const float* __restrict__ thr,
                                                float* __restrict__ M) {
    const int n = blockIdx.x * 256 + threadIdx.x;   // 0..2047
    float acc = 0.0f;
    for (int b = 0; b < BB; ++b) {
        acc += (att [(size_t)b * NN + n] >= thr[b])      ? 0.5f : 0.0f;
        acc += (aoff[(size_t)b * NN + n] >= thr[BB + b]) ? 0.5f : 0.0f;
    }
    M[n] = acc;
}

// ---------------------------------------------------------------------------
// Kernel 3: stable top-SEL selection via rank-by-counting.
// pos(n) = #{m : M[m] > M[n]} + #{m < n : M[m] == M[n]}   (stable argsort(-M))
// Every pos in [0,SEL) is claimed by exactly one n -> deterministic.
// ---------------------------------------------------------------------------
__global__ __launch_bounds__(1024) void rank_kernel(const float* __restrict__ M,
                                                    int*   __restrict__ top_idx,
                                                    float* __restrict__ out_topm) {
    __shared__ float sm[NN];
    const int t = threadIdx.x;
    sm[t]        = M[t];
    sm[t + 1024] = M[t + 1024];
    __syncthreads();

    for (int q = 0; q < 2; ++q) {
        const int   n  = t + q * 1024;
        const float mv = sm[n];
        int pos = 0;
        for (int m = 0; m < NN; ++m) {
            const float o = sm[m];
            pos += (o > mv) ? 1 : 0;
            pos += ((o == mv) && (m < n)) ? 1 : 0;
        }
        if (pos < SEL) {
            top_idx[pos]  = n;
            out_topm[pos] = (float)n;
        }
    }
}

// ---------------------------------------------------------------------------
// Kernel 4: x_sel[b, i, :] = x[b, top_idx[i], :]   (contiguous 16B copies)
// Single-use streaming data -> non-temporal load + store.
// grid = (SEL, BB), block = 128 (DD/4 vectors per row)
// ---------------------------------------------------------------------------
__global__ __launch_bounds__(128) void gx_kernel(const float* __restrict__ x,
                                                 const int*   __restrict__ top_idx,
                                                 float* __restrict__ outx) {
    const int i = blockIdx.x, b = blockIdx.y, t = threadIdx.x;
    const int src = top_idx[i];
    const v4f* s = (const v4f*)(x    + ((size_t)b * NN  + src) * DD);
    v4f*       d = (v4f*)      (outx + ((size_t)b * SEL + i)   * DD);
    v4f v = __builtin_nontemporal_load(&s[t]);
    __builtin_nontemporal_store(v, &d[t]);
}

// ---------------------------------------------------------------------------
// Kernel 5: y_sel[b, i, j] = y[b, top_idx[i], top_idx[j]]
// Stream the full 8KB source row into LDS with CDNA5 async-LDS loads
// (global_load_async_to_lds_b128, ASYNCcnt, NT hint: rows are single-use and
// the touched set (256 MB) exceeds the 192 MB L2), then gather columns
// from LDS and store coalesced. grid = (SEL, BB), block = 256.
// ---------------------------------------------------------------------------
__global__ __launch_bounds__(256) void gy_kernel(const float* __restrict__ y,
                                                 const int*   __restrict__ top_idx,
                                                 float* __restrict__ outy) {
    __shared__ float row[NN];                     // 8 KB staging buffer
    const int i = blockIdx.x, b = blockIdx.y, t = threadIdx.x;
    const int src = top_idx[i];

    const float* g = y + ((size_t)b * NN + src) * NN;
    // Low 32 bits of a generic pointer to LDS are the LDS byte offset
    // (flat LDS address = shared-aperture | offset, ISA §10.2).
    unsigned lds_off = (unsigned)(uintptr_t)(&row[0]) + (unsigned)t * 16u;
    unsigned long long ga = (unsigned long long)(uintptr_t)g
                          + (unsigned long long)t * 16ull;

    // Each lane streams 2x16B: 256 lanes * 32 B = 8192 B = full row.
    // INST_OFFSET applies to both the global and LDS address (ISA async §10).
    asm volatile(
        "global_load_async_to_lds_b128 %0, %1, off th:TH_LOAD_NT\n\t"
        "global_load_async_to_lds_b128 %0, %1, off offset:4096 th:TH_LOAD_NT\n\t"
        "s_wait_asynccnt 0"
        :: "v"(lds_off), "v"(ga) : "memory");
    __syncthreads();

    float* dst = outy + ((size_t)b * SEL + i) * SEL;
    const int j0 = t, j1 = t + 256;
    const float v0 = row[top_idx[j0]];
    const float v1 = row[top_idx[j1]];
    __builtin_nontemporal_store(v0, &dst[j0]);
    __builtin_nontemporal_store(v1, &dst[j1]);
}

// ---------------------------------------------------------------------------
// Launcher
// ---------------------------------------------------------------------------
extern "C" void kernel_launch(void* const* d_in, const int* in_sizes, int n_in,
                              void* d_out, int out_size, void* d_ws, size_t ws_size,
                              hipStream_t stream) {
    const float* x    = (const float*)d_in[0];   // (64, 2048, 512)
    const float* y    = (const float*)d_in[1];   // (64, 2048, 2048)
    const float* att  = (const float*)d_in[2];   // (64, 2048)
    const float* aoff = (const float*)d_in[3];   // (64, 2048)

    float* out  = (float*)d_out;
    float* outx = out;                                        // 64*512*512
    float* outy = out + (size_t)BB * SEL * DD;                // 64*512*512
    float* outm = out + (size_t)2 * BB * SEL * DD;            // 512 floats

    float* ws   = (float*)d_ws;
    float* thr  = ws;                         // 128 floats
    float* M    = ws + 128;                   // 2048 floats
    int*   idx  = (int*)(ws + 128 + NN);      // 512 ints

    thr_kernel <<<2 * BB, 256, 0, stream>>>(att, aoff, thr);
    m_kernel   <<<NN / 256, 256, 0, stream>>>(att, aoff, thr, M);
    rank_kernel<<<1, 1024, 0, stream>>>(M, idx, outm);

    dim3 g(SEL, BB);
    gx_kernel  <<<g, 128, 0, stream>>>(x, idx, outx);
    gy_kernel  <<<g, 256, 0, stream>>>(y, idx, outy);
}